// FinalOneBranchDH_SFNN_86474871538216
// MI455X (gfx1250) — compile-verified
//
#include <hip/hip_runtime.h>
#include <hip/hip_bf16.h>

// SFNN fused pipeline for MI455X (gfx1250, wave32):
//   1) d_in = x @ W_dense^T + b_dense         -- V_WMMA_F32_16X16X4_F32, 16x64 tile/wave
//   2) per-segment zero-init scan             -- massively parallel (B*S*H threads)
//   3) sequential segment combine             -- linear-recurrence homogeneous propagator
//   4) per-segment re-scan + W_out reduce + sigmoid readout
//
// Dims (fixed by the reference):
static constexpr int B_DIM = 32;
static constexpr int T_DIM = 4096;
static constexpr int I_DIM = 128;
static constexpr int H_DIM = 256;
static constexpr int BT    = B_DIM * T_DIM;      // 131072 rows
static constexpr int SEGS  = 32;                 // segments over T
static constexpr int L_SEG = T_DIM / SEGS;       // 128 steps / segment
static constexpr int CT    = 4;                  // column tiles per wave (16x64 tile)

typedef float v2f __attribute__((ext_vector_type(2)));
typedef float v8f __attribute__((ext_vector_type(8)));

__device__ __forceinline__ float sigmoidf_fast(float z) {
    return 1.0f / (1.0f + __expf(-z));
}

// ---------------------------------------------------------------------------
// Kernel 1: d_in[bt, h] = x[bt, :] . W_dense[h, :] + b_dense[h]
// One wave per 16x64 output tile: 4 accumulators sharing one A fragment per
// k-step (5 b64 loads per 4 WMMAs instead of 8). K = 128 -> 32 k-steps,
// 128 WMMA_F32_16X16X4 per wave.
// A fragment (16x4 f32): lanes 0-15 hold (K=k,k+1) for M=lane, lanes 16-31
// hold (K=k+2,k+3) for M=lane-16  => float2 load at rowptr + k + 2*half.
// B fragment (4x16): same K striping, N = lane&15; B[k][n] = W_dense[n][k],
// K-contiguous in W_dense rows => identical float2 addressing off W rows.
// ---------------------------------------------------------------------------
__global__ __launch_bounds__(128) void sfnn_gemm_wmma(
    const float* __restrict__ x, const float* __restrict__ Wd,
    const float* __restrict__ bd, float* __restrict__ din)
{
    const int wave = blockIdx.x * (blockDim.x >> 5) + (threadIdx.x >> 5);
    const int lane = threadIdx.x & 31;
    const int half = lane >> 4;      // 0: lanes 0-15, 1: lanes 16-31
    const int l    = lane & 15;

    const int colTile = wave & (H_DIM / (16 * CT) - 1); // 4 super col tiles
    const int rowTile = wave / (H_DIM / (16 * CT));     // 8192 row tiles
    const int row0 = rowTile * 16;
    const int col0 = colTile * 16 * CT;

    const float* arow = x + (size_t)(row0 + l) * I_DIM + 2 * half;
    const float* brow[CT];
#pragma unroll
    for (int c = 0; c < CT; ++c)
        brow[c] = Wd + (size_t)(col0 + c * 16 + l) * I_DIM + 2 * half;

    // Seed accumulators with bias: C layout has N = lane&15 for every VGPR.
    v8f acc[CT];
#pragma unroll
    for (int c = 0; c < CT; ++c) {
        const float bias = bd[col0 + c * 16 + l];
#pragma unroll
        for (int r = 0; r < 8; ++r) acc[c][r] = bias;
    }

#pragma unroll 8
    for (int k = 0; k < I_DIM; k += 4) {
        const v2f a = *reinterpret_cast<const v2f*>(arow + k);
#pragma unroll
        for (int c = 0; c < CT; ++c) {
            const v2f bf = *reinterpret_cast<const v2f*>(brow[c] + k);
            acc[c] = __builtin_amdgcn_wmma_f32_16x16x4_f32(
                /*neg_a=*/false, a, /*neg_b=*/false, bf,
                /*c_mod=*/(short)0, acc[c], /*reuse_a=*/false, /*reuse_b=*/false);
        }
    }

    // D layout: VGPR r holds row M = r + 8*half, column N = lane&15.
    float* dout = din + (size_t)(row0 + 8 * half) * H_DIM + col0 + l;
#pragma unroll
    for (int c = 0; c < CT; ++c)
#pragma unroll
        for (int r = 0; r < 8; ++r)
            dout[(size_t)r * H_DIM + c * 16] = acc[c][r];
}

// ---------------------------------------------------------------------------
// Kernel 2: per-(b, segment, h) scan from zero state; store (d_end, m_end).
// Consecutive threads map to consecutive h -> fully coalesced din reads.
// ---------------------------------------------------------------------------
__global__ __launch_bounds__(256) void sfnn_seg_scan(
    const float* __restrict__ din, const float* __restrict__ tau_m,
    const float* __restrict__ tau_n, float* __restrict__ seg)
{
    const int tid = blockIdx.x * blockDim.x + threadIdx.x;  // B*SEGS*H
    const int h = tid & (H_DIM - 1);
    const int s = (tid >> 8) & (SEGS - 1);
    const int b = tid >> 13;                                // /(H*SEGS)

    const float alpha = sigmoidf_fast(tau_m[h]);
    const float beta  = sigmoidf_fast(tau_n[h]);

    const float* p = din + ((size_t)b * T_DIM + (size_t)s * L_SEG) * H_DIM + h;
    float d = 0.0f, m = 0.0f;
#pragma unroll 4
    for (int j = 0; j < L_SEG; ++j) {
        const float dt = p[(size_t)j * H_DIM];
        d = beta * d + (1.0f - beta) * dt;
        m = alpha * m + (1.0f - alpha) * d;
    }
    float* q = seg + ((size_t)(b * SEGS + s) * H_DIM + h) * 2;
    q[0] = d;
    q[1] = m;
}

// ---------------------------------------------------------------------------
// Kernel 3: sequential combine over the 32 segments (per b,h chain).
// Homogeneous propagator over L steps:  d_L = beta^L d_0,
//   m_L = alpha^L m_0 + c_dm d_0,  c_dm = sum_{j=1..L} alpha^{L-j}(1-a)beta^j.
// By linearity: trueFinal(s) = Homog(trueInit(s)) + zeroInitFinal(s).
// Overwrites seg[] in place with the TRUE initial state of each segment.
// ---------------------------------------------------------------------------
__global__ __launch_bounds__(256) void sfnn_seg_combine(
    const float* __restrict__ tau_m, const float* __restrict__ tau_n,
    float* __restrict__ seg)
{
    const int tid = blockIdx.x * blockDim.x + threadIdx.x;  // B*H = 8192
    const int h = tid & (H_DIM - 1);
    const int b = tid >> 8;

    const float alpha = sigmoidf_fast(tau_m[h]);
    const float beta  = sigmoidf_fast(tau_n[h]);

    float betaL = 1.0f, c_dm = 0.0f, alphaL = 1.0f;
#pragma unroll 4
    for (int j = 0; j < L_SEG; ++j) {
        betaL *= beta;
        c_dm   = alpha * c_dm + (1.0f - alpha) * betaL;
        alphaL *= alpha;
    }

    float d = 0.0f, m = 0.0f;
    for (int s = 0; s < SEGS; ++s) {
        float* q = seg + ((size_t)(b * SEGS + s) * H_DIM + h) * 2;
        const float ed = q[0];
        const float em = q[1];
        q[0] = d;          // true initial state of segment s
        q[1] = m;
        const float nd = betaL * d + ed;
        const float nm = alphaL * m + c_dm * d + em;
        d = nd;
        m = nm;
    }
}

// ---------------------------------------------------------------------------
// Kernel 4: re-scan each segment from its true initial state, fused with the
// H-reduction against W_out and the sigmoid readout. One block per (b, seg),
// 256 threads = 8 waves, channel h = threadIdx.x. Timesteps processed in
// chunks of 8 to amortize the cross-wave reduction/barriers.
// ---------------------------------------------------------------------------
__global__ __launch_bounds__(256) void sfnn_out_scan(
    const float* __restrict__ din, const float* __restrict__ seg,
    const float* __restrict__ tau_m, const float* __restrict__ tau_n,
    const float* __restrict__ Wout, const float* __restrict__ bout,
    float* __restrict__ out)
{
    const int b  = blockIdx.x >> 5;       // / SEGS
    const int sg = blockIdx.x & 31;
    const int h  = threadIdx.x;
    const int lane = h & 31;
    const int wv   = h >> 5;

    const float alpha = sigmoidf_fast(tau_m[h]);
    const float beta  = sigmoidf_fast(tau_n[h]);
    const float w     = Wout[h];
    const float bo    = bout[0];

    const float* q = seg + ((size_t)(b * SEGS + sg) * H_DIM + h) * 2;
    float d = q[0];
    float m = q[1];

    const float* p = din + ((size_t)b * T_DIM + (size_t)sg * L_SEG) * H_DIM + h;
    float* o = out + (size_t)b * T_DIM + (size_t)sg * L_SEG;

    __shared__ float red[8][8];           // [wave][sub-timestep]

    for (int t0 = 0; t0 < L_SEG; t0 += 8) {
        float c[8];
#pragma unroll
        for (int j = 0; j < 8; ++j) {
            const float dt = p[(size_t)(t0 + j) * H_DIM];
            d = beta * d + (1.0f - beta) * dt;
            m = alpha * m + (1.0f - alpha) * d;
            c[j] = m * w;
        }
        // wave32 butterfly reduction over the 32 channels in this wave
#pragma unroll
        for (int j = 0; j < 8; ++j) {
#pragma unroll
            for (int off = 16; off > 0; off >>= 1)
                c[j] += __shfl_xor(c[j], off, 32);
        }
        if (lane == 0) {
#pragma unroll
            for (int j = 0; j < 8; ++j) red[wv][j] = c[j];
        }
        __syncthreads();
        if (h < 8) {
            float acc = bo;
#pragma unroll
            for (int w2 = 0; w2 < 8; ++w2) acc += red[w2][h];
            o[t0 + h] = sigmoidf_fast(acc);
        }
        __syncthreads();
    }
}

// ---------------------------------------------------------------------------
extern "C" void kernel_launch(void* const* d_in, const int* in_sizes, int n_in,
                              void* d_out, int out_size, void* d_ws, size_t ws_size,
                              hipStream_t stream) {
    const float* x     = (const float*)d_in[0];  // [B,T,I]
    const float* Wd    = (const float*)d_in[1];  // [H,I]
    const float* bd    = (const float*)d_in[2];  // [H]
    const float* tau_m = (const float*)d_in[3];  // [H]
    const float* tau_n = (const float*)d_in[4];  // [H]
    const float* Wout  = (const float*)d_in[5];  // [O=1,H]
    const float* bout  = (const float*)d_in[6];  // [O=1]
    float* out = (float*)d_out;                  // [B,T,1]

    float* din = (float*)d_ws;                          // BT*H floats (128 MB)
    float* seg = din + (size_t)BT * H_DIM;              // B*SEGS*H*2 floats (2 MB)

    // 1) GEMM: one wave per 16x64 tile -> (BT/16)*(H/64) = 32768 waves,
    //    4 waves/block -> 8192 blocks.
    const int waves = (BT / 16) * (H_DIM / (16 * CT));
    sfnn_gemm_wmma<<<waves / 4, 128, 0, stream>>>(x, Wd, bd, din);

    // 2) Per-segment zero-init scan: B*SEGS*H threads.
    sfnn_seg_scan<<<(B_DIM * SEGS * H_DIM) / 256, 256, 0, stream>>>(
        din, tau_m, tau_n, seg);

    // 3) Sequential segment combine: B*H threads.
    sfnn_seg_combine<<<(B_DIM * H_DIM) / 256, 256, 0, stream>>>(
        tau_m, tau_n, seg);

    // 4) Fused re-scan + readout: one block per (b, segment).
    sfnn_out_scan<<<B_DIM * SEGS, 256, 0, stream>>>(
        din, seg, tau_m, tau_n, Wout, bout, out);
}